// CISTransformerEncoder_43336220016867
// MI455X (gfx1250) — compile-verified
//
#include <hip/hip_runtime.h>

typedef __attribute__((ext_vector_type(2))) float v2f;
typedef __attribute__((ext_vector_type(8))) float v8f;

#define L_TOT 13294
#define DM 256

#if __has_builtin(__builtin_amdgcn_s_wait_asynccnt)
#define ASYNC_WAIT(n) __builtin_amdgcn_s_wait_asynccnt(n)
#else
#define ASYNC_WAIT(n) asm volatile("s_wait_asynccnt %0" ::"i"(n) : "memory")
#endif

// ---------------------------------------------------------------------------
// WMMA GEMM:  C[M,N] = A[M,K] @ W[K,N] + bias, optional ReLU.
// Block = 256 threads (8 waves, arranged 4(M) x 2(N)).
// Block tile BM=64 x BN=128; each wave owns a 16x64 output (4 x v8f acc).
// K staged in BK=32 chunks, double-buffered in LDS, filled with
// global_load_async_to_lds_b128 (ASYNCcnt) so tile i+1 loads overlap tile i
// WMMA compute.
// ---------------------------------------------------------------------------
#define BM 64
#define BN 128
#define BK 32

__device__ __forceinline__ void issue_tile(
    const float* __restrict__ A, const float* __restrict__ W,
    float* AsBuf, float* BsBuf,
    int m0, int n0, int k0, int M, int N, int K, int tid)
{
    const unsigned long long Abase = (unsigned long long)A;
    const unsigned long long Wbase = (unsigned long long)W;
    // A tile: 64 x 32 floats = 512 float4 -> 2 async b128 per thread
#pragma unroll
    for (int j = 0; j < 2; ++j) {
        int idx = tid + j * 256;
        int row = idx >> 3;             // 0..63
        int c4  = (idx & 7) << 2;       // 0..28
        int gr  = m0 + row; if (gr >= M) gr = M - 1;   // clamp: async reads are unconditional
        unsigned voff = (unsigned)(((size_t)gr * K + k0 + c4) * 4u);
        unsigned lds  = (unsigned)(uintptr_t)(AsBuf + row * (BK + 4) + c4);
        asm volatile("global_load_async_to_lds_b128 %0, %1, %2"
                     :: "v"(lds), "v"(voff), "s"(Abase) : "memory");
    }
    // B tile: 32 x 128 floats = 1024 float4 -> 4 async b128 per thread
#pragma unroll
    for (int j = 0; j < 4; ++j) {
        int idx = tid + j * 256;
        int row = idx >> 5;             // 0..31
        int c4  = (idx & 31) << 2;      // 0..124
        unsigned voff = (unsigned)(((size_t)(k0 + row) * N + n0 + c4) * 4u);
        unsigned lds  = (unsigned)(uintptr_t)(BsBuf + row * (BN + 4) + c4);
        asm volatile("global_load_async_to_lds_b128 %0, %1, %2"
                     :: "v"(lds), "v"(voff), "s"(Wbase) : "memory");
    }
}

__global__ __launch_bounds__(256) void gemm_wmma_f32(
    const float* __restrict__ A, const float* __restrict__ W,
    const float* __restrict__ bias, float* __restrict__ C,
    int M, int N, int K, int relu)
{
    __shared__ float As[2][BM][BK + 4];   // 2 * 64 * 36 * 4B = 18 KB
    __shared__ float Bs[2][BK][BN + 4];   // 2 * 32 * 132 * 4B = 33 KB

    const int tid  = threadIdx.x;
    const int lane = tid & 31;
    const int wid  = tid >> 5;     // 0..7
    const int wm   = wid >> 1;     // 0..3  (M sub-tile)
    const int wn   = wid & 1;      // 0..1  (N sub-tile, 64 cols each)
    const int m0   = blockIdx.x * BM;
    const int n0   = blockIdx.y * BN;

    const int lr = lane & 15;      // 0..15
    const int lh = lane >> 4;      // 0..1

    // C/D layout: VGPR r, lane -> row = r + (lane/16)*8, col = lane%16
    v8f acc[4];
#pragma unroll
    for (int t = 0; t < 4; ++t) {
        const float bv = bias[n0 + wn * 64 + t * 16 + lr];
#pragma unroll
        for (int r = 0; r < 8; ++r) acc[t][r] = bv;
    }

    const int ntiles = K / BK;
    issue_tile(A, W, &As[0][0][0], &Bs[0][0][0], m0, n0, 0, M, N, K, tid);

    for (int it = 0; it < ntiles; ++it) {
        const int cur = it & 1;
        if (it + 1 < ntiles) {
            issue_tile(A, W, &As[cur ^ 1][0][0], &Bs[cur ^ 1][0][0],
                       m0, n0, (it + 1) * BK, M, N, K, tid);
            ASYNC_WAIT(6);   // 6 asyncs of tile it+1 in flight; tile it retired
        } else {
            ASYNC_WAIT(0);
        }
        __syncthreads();     // all waves' portions of tile `cur` are in LDS

#pragma unroll
        for (int ks = 0; ks < BK; ks += 4) {
            const int ka = ks + lh * 2;   // A/B lane K-offset per ISA layout
            float2 af = *(const float2*)&As[cur][wm * 16 + lr][ka];
            v2f a; a[0] = af.x; a[1] = af.y;
#pragma unroll
            for (int t = 0; t < 4; ++t) {
                v2f b;
                b[0] = Bs[cur][ka][wn * 64 + t * 16 + lr];
                b[1] = Bs[cur][ka + 1][wn * 64 + t * 16 + lr];
                acc[t] = __builtin_amdgcn_wmma_f32_16x16x4_f32(
                    false, a, false, b, (short)0, acc[t], false, false);
            }
        }
        __syncthreads();     // done reading buf `cur` before it is refilled
    }

#pragma unroll
    for (int t = 0; t < 4; ++t) {
#pragma unroll
        for (int r = 0; r < 8; ++r) {
            int m = m0 + wm * 16 + r + lh * 8;
            if (m < M) {
                float x = acc[t][r];
                if (relu) x = fmaxf(x, 0.f);
                C[(size_t)m * N + n0 + wn * 64 + t * 16 + lr] = x;
            }
        }
    }
}

// ---------------------------------------------------------------------------
// q = a + b (elementwise)
// ---------------------------------------------------------------------------
__global__ void add_kernel(float* __restrict__ q, const float* __restrict__ a,
                           const float* __restrict__ b, int total)
{
    int t = blockIdx.x * 256 + threadIdx.x;
    if (t < total) q[t] = a[t] + b[t];
}

// ---------------------------------------------------------------------------
// Flatten src/pos (B,D,H,W) -> (B, L, D); pos += level_embed.
// ---------------------------------------------------------------------------
__global__ void flatten_kernel(
    const float* __restrict__ s0, const float* __restrict__ s1,
    const float* __restrict__ s2, const float* __restrict__ s3,
    const float* __restrict__ p0, const float* __restrict__ p1,
    const float* __restrict__ p2, const float* __restrict__ p3,
    const float* __restrict__ lev, float* __restrict__ src,
    float* __restrict__ pos, int total)
{
    int t = blockIdx.x * 256 + threadIdx.x;
    if (t >= total) return;
    int d = t & (DM - 1);
    int m = t >> 8;
    int b = m / L_TOT;
    int q = m % L_TOT;
    int l, idx, HW; const float *sp, *pp;
    if (q < 10000)      { l = 0; idx = q;         HW = 10000; sp = s0; pp = p0; }
    else if (q < 12500) { l = 1; idx = q - 10000; HW = 2500;  sp = s1; pp = p1; }
    else if (q < 13125) { l = 2; idx = q - 12500; HW = 625;   sp = s2; pp = p2; }
    else                { l = 3; idx = q - 13125; HW = 169;   sp = s3; pp = p3; }
    size_t gi = ((size_t)b * DM + d) * HW + idx;
    src[t] = sp[gi];
    pos[t] = pp[gi] + lev[l * DM + d];
}

// ---------------------------------------------------------------------------
// Reference points per token (same across NL in the reference).
// ---------------------------------------------------------------------------
__global__ void ref_kernel(float* __restrict__ ref, int L)
{
    int q = blockIdx.x * 256 + threadIdx.x;
    if (q >= L) return;
    int idx, H, W;
    if (q < 10000)      { idx = q;         H = 100; W = 100; }
    else if (q < 12500) { idx = q - 10000; H = 50;  W = 50;  }
    else if (q < 13125) { idx = q - 12500; H = 25;  W = 25;  }
    else                { idx = q - 13125; H = 13;  W = 13;  }
    int row = idx / W, col = idx % W;
    ref[2 * q]     = (col + 0.5f) / (float)W;
    ref[2 * q + 1] = (row + 0.5f) / (float)H;
}

__global__ void tail_kernel(float* __restrict__ out, int M)
{
    if (threadIdx.x < 4) {
        const int st[4] = {0, 10000, 12500, 13125};
        ((int*)(out + (size_t)M * DM))[threadIdx.x] = st[threadIdx.x];
    }
}

// ---------------------------------------------------------------------------
// Softmax over 16 contiguous attention weights per (b, q, head).
// ---------------------------------------------------------------------------
__global__ void softmax16_kernel(float* __restrict__ aw, int rows)
{
    int i = blockIdx.x * 256 + threadIdx.x;
    if (i >= rows) return;
    float* p = aw + (size_t)i * 16;
    float v[16], mx = -3.4e38f;
#pragma unroll
    for (int j = 0; j < 16; ++j) { v[j] = p[j]; mx = fmaxf(mx, v[j]); }
    float s = 0.f;
#pragma unroll
    for (int j = 0; j < 16; ++j) { v[j] = __expf(v[j] - mx); s += v[j]; }
    float inv = 1.f / s;
#pragma unroll
    for (int j = 0; j < 16; ++j) p[j] = v[j] * inv;
}

// ---------------------------------------------------------------------------
// Multi-scale deformable sampling: one wave32 per (b, q, head); lane = channel.
// ---------------------------------------------------------------------------
__global__ __launch_bounds__(256) void sample_kernel(
    const float* __restrict__ value, const float* __restrict__ offs,
    const float* __restrict__ aw, const float* __restrict__ ref,
    float* __restrict__ out, int nwaves, int L)
{
    int gw = blockIdx.x * 8 + (threadIdx.x >> 5);
    if (gw >= nwaves) return;
    int lane = threadIdx.x & 31;
    int h = gw & 7;
    int m = gw >> 3;            // b*L + q
    int q = m % L;
    float rx = ref[2 * q], ry = ref[2 * q + 1];
    const float* op  = offs + (size_t)m * DM + h * 32;
    const float* awp = aw   + (size_t)m * 128 + h * 16;

    const int Hs[4] = {100, 50, 25, 13};
    const int st[4] = {0, 10000, 12500, 13125};
    float acc = 0.f;
#pragma unroll
    for (int l = 0; l < 4; ++l) {
        const int Hl = Hs[l], Wl = Hs[l];
        const float* vb = value + ((size_t)(m - q) + (size_t)st[l]) * DM + h * 32 + lane;
#pragma unroll
        for (int p = 0; p < 4; ++p) {
            float px = (rx + op[l * 8 + p * 2 + 0] / (float)Wl) * Wl - 0.5f;
            float py = (ry + op[l * 8 + p * 2 + 1] / (float)Hl) * Hl - 0.5f;
            float x0 = floorf(px), y0 = floorf(py);
            float wx1 = px - x0, wy1 = py - y0;
            int xi0 = (int)x0, yi0 = (int)y0;
            float s = 0.f;
#pragma unroll
            for (int dy = 0; dy < 2; ++dy) {
#pragma unroll
                for (int dx = 0; dx < 2; ++dx) {
                    int xi = xi0 + dx, yi = yi0 + dy;
                    if (xi >= 0 && xi < Wl && yi >= 0 && yi < Hl) {
                        float w = (dx ? wx1 : 1.f - wx1) * (dy ? wy1 : 1.f - wy1);
                        s += w * vb[(size_t)(yi * Wl + xi) * DM];
                    }
                }
            }
            acc += awp[l * 4 + p] * s;
        }
    }
    out[(size_t)m * DM + h * 32 + lane] = acc;
}

// ---------------------------------------------------------------------------
// src = LayerNorm(src + res) * w + b ; one block (256 thr) per row.
// ---------------------------------------------------------------------------
__global__ __launch_bounds__(256) void add_ln_kernel(
    float* __restrict__ src, const float* __restrict__ res,
    const float* __restrict__ w, const float* __restrict__ b)
{
    __shared__ float red[8];
    int row = blockIdx.x, d = threadIdx.x;
    int lane = d & 31, wid = d >> 5;
    float x = src[(size_t)row * DM + d] + res[(size_t)row * DM + d];

    float s = x;
#pragma unroll
    for (int o = 16; o; o >>= 1) s += __shfl_xor(s, o, 32);
    if (lane == 0) red[wid] = s;
    __syncthreads();
    float tot = 0.f;
#pragma unroll
    for (int i = 0; i < 8; ++i) tot += red[i];
    float mean = tot * (1.f / DM);
    float dx = x - mean;
    __syncthreads();

    float s2 = dx * dx;
#pragma unroll
    for (int o = 16; o; o >>= 1) s2 += __shfl_xor(s2, o, 32);
    if (lane == 0) red[wid] = s2;
    __syncthreads();
    float vtot = 0.f;
#pragma unroll
    for (int i = 0; i < 8; ++i) vtot += red[i];
    float var = vtot * (1.f / DM);

    src[(size_t)row * DM + d] = dx * rsqrtf(var + 1e-5f) * w[d] + b[d];
}

// ---------------------------------------------------------------------------
static inline void launch_gemm(const float* A, const float* W, const float* bias,
                               float* C, int M, int N, int K, int relu,
                               hipStream_t st)
{
    dim3 grid((M + BM - 1) / BM, N / BN);
    gemm_wmma_f32<<<grid, 256, 0, st>>>(A, W, bias, C, M, N, K, relu);
}

extern "C" void kernel_launch(void* const* d_in, const int* in_sizes, int n_in,
                              void* d_out, int out_size, void* d_ws, size_t ws_size,
                              hipStream_t stream)
{
    (void)in_sizes; (void)n_in; (void)out_size; (void)ws_size;
    const float* s0 = (const float*)d_in[0];
    const float* s1 = (const float*)d_in[1];
    const float* s2 = (const float*)d_in[2];
    const float* s3 = (const float*)d_in[3];
    const float* p0 = (const float*)d_in[4];
    const float* p1 = (const float*)d_in[5];
    const float* p2 = (const float*)d_in[6];
    const float* p3 = (const float*)d_in[7];
    const float* lev  = (const float*)d_in[8];
    const float* so_w = (const float*)d_in[9];
    const float* so_b = (const float*)d_in[10];
    const float* aw_w = (const float*)d_in[11];
    const float* aw_b = (const float*)d_in[12];
    const float* vp_w = (const float*)d_in[13];
    const float* vp_b = (const float*)d_in[14];
    const float* op_w = (const float*)d_in[15];
    const float* op_b = (const float*)d_in[16];
    const float* n1_w = (const float*)d_in[17];
    const float* n1_b = (const float*)d_in[18];
    const float* f1_w = (const float*)d_in[19];
    const float* f1_b = (const float*)d_in[20];
    const float* f2_w = (const float*)d_in[21];
    const float* f2_b = (const float*)d_in[22];
    const float* n2_w = (const float*)d_in[23];
    const float* n2_b = (const float*)d_in[24];

    const int L = L_TOT, M = 4 * L_TOT;   // 53176 rows
    float* src = (float*)d_out;           // evolving activation lives in d_out
    float* ws  = (float*)d_ws;
    const size_t MF = (size_t)M * DM;     // 13,613,056 floats
    float* pos   = ws;
    float* value = ws + MF;
    float* offs  = ws + 2 * MF;
    float* attn  = ws + 3 * MF;           // reused as FFN hidden chunk buffer
    float* proj  = ws + 4 * MF;           // also holds q = src + pos early in layer
    float* awb   = ws + 5 * MF;           // M*128
    float* refp  = ws + 5 * MF + (size_t)M * 128;

    const int total = M * DM;
    flatten_kernel<<<(total + 255) / 256, 256, 0, stream>>>(
        s0, s1, s2, s3, p0, p1, p2, p3, lev, src, pos, total);
    ref_kernel<<<(L + 255) / 256, 256, 0, stream>>>(refp, L);
    tail_kernel<<<1, 32, 0, stream>>>(src, M);

    const int MC = M / 4;                 // 13294 (exact); MC*1024 == MF
    for (int i = 0; i < 6; ++i) {
        // value projection
        launch_gemm(src, vp_w + (size_t)i * 256 * 256, vp_b + i * 256,
                    value, M, 256, 256, 0, stream);
        // q = src + pos (into proj buffer, free until output projection)
        add_kernel<<<(total + 255) / 256, 256, 0, stream>>>(proj, src, pos, total);
        // sampling offsets + attention weights from q
        launch_gemm(proj, so_w + (size_t)i * 256 * 256, so_b + i * 256,
                    offs, M, 256, 256, 0, stream);
        launch_gemm(proj, aw_w + (size_t)i * 256 * 128, aw_b + i * 128,
                    awb, M, 128, 256, 0, stream);
        softmax16_kernel<<<(M * 8 + 255) / 256, 256, 0, stream>>>(awb, M * 8);
        sample_kernel<<<(M * 8 + 7) / 8, 256, 0, stream>>>(
            value, offs, awb, refp, attn, M * 8, L);
        // output projection (overwrites q; q is dead by now)
        launch_gemm(attn, op_w + (size_t)i * 256 * 256, op_b + i * 256,
                    proj, M, 256, 256, 0, stream);
        add_ln_kernel<<<M, 256, 0, stream>>>(src, proj, n1_w + i * 256, n1_b + i * 256);

        // FFN, chunked 4x so hidden fits the reused 'attn' buffer exactly
        for (int c = 0; c < 4; ++c) {
            const size_t r0 = (size_t)c * MC;
            launch_gemm(src + r0 * 256, f1_w + (size_t)i * 256 * 1024,
                        f1_b + i * 1024, attn, MC, 1024, 256, 1, stream);
            launch_gemm(attn, f2_w + (size_t)i * 1024 * 256,
                        f2_b + i * 256, proj + r0 * 256, MC, 256, 1024, 0, stream);
        }
        add_ln_kernel<<<M, 256, 0, stream>>>(src, proj, n2_w + i * 256, n2_b + i * 256);
    }
}